// STGraphConstructor_63393717289324
// MI455X (gfx1250) — compile-verified
//
#include <hip/hip_runtime.h>
#include <hip/hip_bf16.h>

typedef __attribute__((ext_vector_type(2))) float v2f;
typedef __attribute__((ext_vector_type(8))) float v8f;

#define BATCH 16
#define NS    2048
#define NT    168
#define MTOT  (NS + NT)            // 2216
#define DDIM  64
#define TILES ((MTOT + 15) / 16)   // 139
#define WAVES_PER_BLOCK 4

__device__ __forceinline__ const float* row_ptr(const float* __restrict__ sp,
                                                const float* __restrict__ tp,
                                                int b, int r) {
    // r < MTOT guaranteed by caller (clamped)
    if (r < NS) return sp + ((size_t)b * NS + r) * DDIM;
    return tp + ((size_t)b * NT + (r - NS)) * DDIM;
}

// tanh(relu(x)) without branches: for r = max(x,0),
// tanh(r) = (1 - e) / (1 + e), e = exp(-2r) in (0,1]  -> stable, no overflow.
__device__ __forceinline__ float tanh_relu(float x) {
    float r = fmaxf(x, 0.0f);
    float e = __expf(-2.0f * r);                       // v_exp_f32 path
    return (1.0f - e) * __builtin_amdgcn_rcpf(1.0f + e);
}

__global__ __launch_bounds__(WAVES_PER_BLOCK * 32)
void stgraph_wmma_f32(const float* __restrict__ sp,
                      const float* __restrict__ tp,
                      float* __restrict__ out) {
    const int lane = threadIdx.x & 31;
    const int wave = threadIdx.x >> 5;
    const int tj   = blockIdx.x * WAVES_PER_BLOCK + wave;  // wave-uniform
    const int ti   = blockIdx.y;
    const int b    = blockIdx.z;
    if (tj >= TILES) return;  // uniform per wave -> EXEC all-1s at WMMA

    const int i0   = ti * 16;
    const int j0   = tj * 16;
    const int half = lane >> 4;   // 0: K = {0,1}+4s ; 1: K = {2,3}+4s
    const int lrow = lane & 15;

    // Clamp edge rows: loads stay in-bounds; out-of-range results never stored.
    int arow = i0 + lrow; if (arow >= MTOT) arow = MTOT - 1;
    int brow = j0 + lrow; if (brow >= MTOT) brow = MTOT - 1;

    const float* __restrict__ aptr = row_ptr(sp, tp, b, arow) + half * 2;
    const float* __restrict__ bptr = row_ptr(sp, tp, b, brow) + half * 2;

    v8f c = {};
#pragma unroll
    for (int s = 0; s < DDIM / 4; ++s) {
        v2f av = *(const v2f*)(aptr + s * 4);   // X[arow][4s + 2*half + {0,1}]
        v2f bv = *(const v2f*)(bptr + s * 4);   // X[brow][4s + 2*half + {0,1}]
        c = __builtin_amdgcn_wmma_f32_16x16x4_f32(
                /*neg_a=*/false, av, /*neg_b=*/false, bv,
                /*c_mod=*/(short)0, c, /*reuse_a=*/false, /*reuse_b=*/false);
    }

    // Epilogue: tanh(relu(x)), coalesced strided stores.
    const size_t M = (size_t)MTOT;
    const int n = j0 + lrow;

    if ((i0 + 16 <= MTOT) && (j0 + 16 <= MTOT)) {
        // Interior tile (common case): no per-element guards.
        float* __restrict__ p =
            out + ((size_t)b * M + (size_t)(i0 + 8 * half)) * M + (size_t)n;
#pragma unroll
        for (int r = 0; r < 8; ++r) {
            p[(size_t)r * M] = tanh_relu(c[r]);
        }
    } else {
        // Edge tile: guard each element.
#pragma unroll
        for (int r = 0; r < 8; ++r) {
            const int m = i0 + r + 8 * half;
            if (m < MTOT && n < MTOT) {
                out[((size_t)b * M + (size_t)m) * M + (size_t)n] = tanh_relu(c[r]);
            }
        }
    }
}

extern "C" void kernel_launch(void* const* d_in, const int* in_sizes, int n_in,
                              void* d_out, int out_size, void* d_ws, size_t ws_size,
                              hipStream_t stream) {
    (void)in_sizes; (void)n_in; (void)out_size; (void)d_ws; (void)ws_size;
    const float* sp = (const float*)d_in[0];  // [16, 2048, 64] fp32
    const float* tp = (const float*)d_in[1];  // [16, 168, 64] fp32
    float* out = (float*)d_out;               // [16, 2216, 2216] fp32

    dim3 grid((TILES + WAVES_PER_BLOCK - 1) / WAVES_PER_BLOCK, TILES, BATCH);
    dim3 block(WAVES_PER_BLOCK * 32, 1, 1);
    stgraph_wmma_f32<<<grid, block, 0, stream>>>(sp, tp, out);
}